// CRD_2310692405648
// MI455X (gfx1250) — compile-verified
//
#include <hip/hip_runtime.h>

#define D_IN  128
#define D_OUT 64

typedef __attribute__((ext_vector_type(2))) float v2f;
typedef __attribute__((ext_vector_type(8))) float v8f;

// ---- degree init: every node gets a self-loop ----
__global__ __launch_bounds__(256) void k_init_deg(int* __restrict__ deg, int n) {
  int i = blockIdx.x * blockDim.x + threadIdx.x;
  if (i < n) deg[i] = 1;
}

// ---- degree count over edge targets (row = edge_index[0]) ----
__global__ __launch_bounds__(256) void k_count(const int* __restrict__ ei,
                                               int* __restrict__ deg, int e) {
  int idx = blockIdx.x * blockDim.x + threadIdx.x;
  if (idx < e) {
    int t = __builtin_nontemporal_load(ei + idx);  // streaming: keep L2 for h/out
    atomicAdd(deg + t, 1);
  }
}

// ---- dis[i] = rsqrt(deg[i])  (deg >= 1 always, self-loops) ----
__global__ __launch_bounds__(256) void k_dis(const int* __restrict__ deg,
                                             float* __restrict__ dis, int n) {
  int i = blockIdx.x * blockDim.x + threadIdx.x;
  if (i < n) dis[i] = rsqrtf((float)deg[i]);
}

// ---- zero accumulator (d_out is poisoned, not re-zeroed by harness) ----
__global__ __launch_bounds__(256) void k_zero(float* __restrict__ out, long long total) {
  long long i = (long long)blockIdx.x * blockDim.x + threadIdx.x;
  if (i < total) out[i] = 0.0f;
}

// ---- h = x @ W via V_WMMA_F32_16X16X4_F32, one wave per 16x16 tile ----
__global__ __launch_bounds__(256) void k_gemm_wmma(const float* __restrict__ x,
                                                   const float* __restrict__ Wm,
                                                   float* __restrict__ h, int n) {
  const int lane  = threadIdx.x & 31;
  const int wave  = blockIdx.x * (blockDim.x >> 5) + (threadIdx.x >> 5);
  const int ntiles = D_OUT / 16;                 // 4
  const int mtiles = (n + 15) >> 4;
  const int mtile = wave / ntiles;
  const int ntile = wave % ntiles;
  if (mtile >= mtiles) return;                   // wave-uniform: whole wave exits

  // A fragment addressing: lane -> row M = lane%16, K-half = 2*(lane/16)
  const int mrow  = mtile * 16 + (lane & 15);
  const int m     = (mrow < n) ? mrow : (n - 1); // clamp loads on tail tile
  const int khalf = (lane >> 4) << 1;            // 0 or 2
  const int ncol  = ntile * 16 + (lane & 15);    // B/D column = lane%16

  const float* arow = x + (size_t)m * D_IN;
  v8f c = {};
#pragma unroll
  for (int k = 0; k < D_IN; k += 4) {
    v2f a, b;
    a.x = arow[k + khalf];
    a.y = arow[k + khalf + 1];
    b.x = Wm[(size_t)(k + khalf) * D_OUT + ncol];
    b.y = Wm[(size_t)(k + khalf + 1) * D_OUT + ncol];
    // 8 args: (neg_a, A, neg_b, B, c_mod, C, reuse_a, reuse_b)
    c = __builtin_amdgcn_wmma_f32_16x16x4_f32(false, a, false, b, (short)0, c,
                                              false, false);
  }
  // D layout: VGPR r holds M = r (lanes 0-15) / M = r+8 (lanes 16-31)
  const int mbase = mtile * 16 + ((lane >> 4) << 3);
#pragma unroll
  for (int r = 0; r < 8; ++r) {
    int row = mbase + r;
    if (row < n) h[(size_t)row * D_OUT + ncol] = c[r];
  }
}

// ---- edge scatter: out[row] += dis[row]*dis[col] * h[col]; 16 lanes/edge ----
__global__ __launch_bounds__(256) void k_scatter(const int* __restrict__ ei,
                                                 const float* __restrict__ h,
                                                 const float* __restrict__ dis,
                                                 float* __restrict__ out, int e) {
  long long gid  = (long long)blockIdx.x * blockDim.x + threadIdx.x;
  long long eidx = gid >> 4;
  if (eidx >= e) return;
  int r = (int)(gid & 15);
  int t = __builtin_nontemporal_load(ei + eidx);               // target (row)
  int s = __builtin_nontemporal_load(ei + (long long)e + eidx); // source (col)
  float w = dis[t] * dis[s];
  const float4* hv = (const float4*)(h + (size_t)s * D_OUT);
  float4 v = hv[r];                                            // L2-resident gather
  float* op = out + (size_t)t * D_OUT + (r << 2);
  atomicAdd(op + 0, w * v.x);   // global_atomic_add_f32, serviced in L2
  atomicAdd(op + 1, w * v.y);
  atomicAdd(op + 2, w * v.z);
  atomicAdd(op + 3, w * v.w);
}

// ---- self-loop term + bias + ReLU, in place ----
__global__ __launch_bounds__(256) void k_finalize(const float* __restrict__ h,
                                                  const float* __restrict__ dis,
                                                  const float* __restrict__ bias,
                                                  float* __restrict__ out,
                                                  long long total) {
  long long gid = (long long)blockIdx.x * blockDim.x + threadIdx.x;
  if (gid >= total) return;
  int i = (int)(gid >> 6);           // node
  int d = (int)(gid & (D_OUT - 1));  // feature
  float di = dis[i];
  float v = out[gid] + di * di * h[gid] + bias[d];
  out[gid] = v > 0.0f ? v : 0.0f;
}

extern "C" void kernel_launch(void* const* d_in, const int* in_sizes, int n_in,
                              void* d_out, int out_size, void* d_ws, size_t ws_size,
                              hipStream_t stream) {
  const float* x  = (const float*)d_in[0];  // [N, 128]
  const int*   ei = (const int*)d_in[1];    // [2, E] flattened, row-major
  const float* Wm = (const float*)d_in[2];  // [128, 64]
  const float* b  = (const float*)d_in[3];  // [64]
  const int n = in_sizes[0] / D_IN;
  const int e = in_sizes[1] / 2;
  float* out = (float*)d_out;

  // workspace: deg[int,N] | dis[f32,N] | h[f32,N*64]  (~26.5 MB)
  int*   deg = (int*)d_ws;
  float* dis = (float*)d_ws + n;
  float* h   = (float*)d_ws + 2 * (size_t)n;

  const long long total = (long long)n * D_OUT;

  k_init_deg<<<(n + 255) / 256, 256, 0, stream>>>(deg, n);
  k_count<<<(e + 255) / 256, 256, 0, stream>>>(ei, deg, e);
  k_dis<<<(n + 255) / 256, 256, 0, stream>>>(deg, dis, n);
  k_zero<<<(unsigned)((total + 255) / 256), 256, 0, stream>>>(out, total);

  const int mtiles = (n + 15) / 16;
  const int waves  = mtiles * (D_OUT / 16);
  k_gemm_wmma<<<(waves + 7) / 8, 256, 0, stream>>>(x, Wm, h, n);

  const long long sthreads = (long long)e * 16;
  k_scatter<<<(unsigned)((sthreads + 255) / 256), 256, 0, stream>>>(ei, h, dis, out, e);

  k_finalize<<<(unsigned)((total + 255) / 256), 256, 0, stream>>>(h, dis, b, out, total);
}